// TensorProduct_60851096649887
// MI455X (gfx1250) — compile-verified
//
#include <hip/hip_runtime.h>

typedef __attribute__((ext_vector_type(2))) float v2f;
typedef __attribute__((ext_vector_type(8))) float v8f;

#define MULC 256
#define MT 32          // samples per workgroup
#define ASTRIDE 260    // padded LDS row stride (floats): 8B-aligned pairs, conflict-free banks

__global__ __launch_bounds__(512)
void cg_tp_wmma_f32(const float* __restrict__ x1,
                    const float* __restrict__ x2,
                    const float* __restrict__ Wt,
                    float* __restrict__ out,
                    int n_total) {
    // Six A matrices, 32 rows x 256 cols each, scalar factors pre-folded. ~200 KB LDS.
    __shared__ float A[6][MT][ASTRIDE];
    __shared__ float XV[MT][4];   // per-sample [s0, v0, v1, v2]

    const int tid = threadIdx.x;
    const int n0 = blockIdx.x * MT;
    if (n0 >= n_total) return;

    const float isq3 = 0.57735026918962576451f;

    // ---- Phase 1: build A0..A5 in LDS ----
    if (tid < MT * 4) {
        int m = tid >> 2;
        int nrow = n0 + m; if (nrow >= n_total) nrow = n_total - 1;
        XV[m][tid & 3] = x2[(size_t)nrow * 4 + (tid & 3)];
    }
    __syncthreads();

    {
        const int u = tid & 255;           // 256 columns, coalesced row reads
        const int mbase = (tid >> 8) * 16; // threads 0-255 -> rows 0..15, 256-511 -> 16..31
        for (int mm = 0; mm < 16; ++mm) {
            const int m = mbase + mm;
            int nrow = n0 + m; if (nrow >= n_total) nrow = n_total - 1;  // clamp (stores guarded)
            const float* row = x1 + (size_t)nrow * 1024;
            float av = row[u];
            float b0 = row[256 + u];
            float b1 = row[512 + u];
            float b2 = row[768 + u];
            float s0 = XV[m][0], v0 = XV[m][1], v1 = XV[m][2], v2 = XV[m][3];
            A[0][m][u] = av * s0;                               // -> W0
            A[1][m][u] = isq3 * (v0 * b0 + v1 * b1 + v2 * b2);  // -> W3
            A[2][m][u] = isq3 * av;                             // -> W1
            A[3][m][u] = isq3 * s0 * b0;                        // -> W2
            A[4][m][u] = isq3 * s0 * b1;                        // -> W2
            A[5][m][u] = isq3 * s0 * b2;                        // -> W2
        }
    }
    __syncthreads();

    // ---- Phase 2: WMMA fp32 GEMMs ----
    // 16 waves: wave wv owns one 16-column tile and BOTH 16-row m-tiles,
    // so B fragments (weights) are loaded once per k-step and reused 2x.
    const int lane = tid & 31;
    const int wv   = tid >> 5;       // 0..15 -> column tile
    const int h    = lane >> 4;      // half-wave select
    const int ml   = lane & 15;      // M row within A fragment / N col within B,C,D
    const int c    = wv * 16 + ml;   // this lane's output column (within 256)

    const float* W0 = Wt;
    const float* W1 = Wt + 1 * MULC * MULC;
    const float* W2 = Wt + 2 * MULC * MULC;
    const float* W3 = Wt + 3 * MULC * MULC;

    v8f acc[2][6] = {};              // [m-tile][path-tile], 96 VGPRs of fp32 accum

    #pragma unroll 2
    for (int k = 0; k < MULC; k += 4) {
        const int kk = k + 2 * h;    // VGPR0 <-> K=kk, VGPR1 <-> K=kk+1 per ISA layout

        const size_t r0 = (size_t)kk * MULC;
        const size_t r1 = r0 + MULC;
        v2f bw0 = { W0[r0 + c], W0[r1 + c] };
        v2f bw1 = { W1[r0 + c], W1[r1 + c] };
        v2f bw2 = { W2[r0 + c], W2[r1 + c] };
        v2f bw3 = { W3[r0 + c], W3[r1 + c] };

        #pragma unroll
        for (int mt = 0; mt < 2; ++mt) {
            const int mr = mt * 16 + ml;
            v2f fa0 = *(const v2f*)&A[0][mr][kk];
            v2f fa1 = *(const v2f*)&A[1][mr][kk];
            v2f fa2 = *(const v2f*)&A[2][mr][kk];
            v2f fa3 = *(const v2f*)&A[3][mr][kk];
            v2f fa4 = *(const v2f*)&A[4][mr][kk];
            v2f fa5 = *(const v2f*)&A[5][mr][kk];

            acc[mt][0] = __builtin_amdgcn_wmma_f32_16x16x4_f32(false, fa0, false, bw0, (short)0, acc[mt][0], false, false);
            acc[mt][1] = __builtin_amdgcn_wmma_f32_16x16x4_f32(false, fa1, false, bw3, (short)0, acc[mt][1], false, false);
            acc[mt][2] = __builtin_amdgcn_wmma_f32_16x16x4_f32(false, fa2, false, bw1, (short)0, acc[mt][2], false, false);
            acc[mt][3] = __builtin_amdgcn_wmma_f32_16x16x4_f32(false, fa3, false, bw2, (short)0, acc[mt][3], false, false);
            acc[mt][4] = __builtin_amdgcn_wmma_f32_16x16x4_f32(false, fa4, false, bw2, (short)0, acc[mt][4], false, false);
            acc[mt][5] = __builtin_amdgcn_wmma_f32_16x16x4_f32(false, fa5, false, bw2, (short)0, acc[mt][5], false, false);
        }
    }

    // ---- Epilogue: combine paths, store (C/D layout: VGPR r -> M = r + 8*h) ----
    #pragma unroll
    for (int mt = 0; mt < 2; ++mt) {
        #pragma unroll
        for (int r = 0; r < 8; ++r) {
            const int m = mt * 16 + r + 8 * h;
            const int nrow = n0 + m;
            if (nrow < n_total) {
                const size_t rowbase = (size_t)nrow * 1024;
                const float v0 = XV[m][1], v1 = XV[m][2], v2 = XV[m][3];
                out[rowbase + c]       = acc[mt][0][r] + acc[mt][1][r];
                out[rowbase + 256 + c] = v0 * acc[mt][2][r] + acc[mt][3][r];
                out[rowbase + 512 + c] = v1 * acc[mt][2][r] + acc[mt][4][r];
                out[rowbase + 768 + c] = v2 * acc[mt][2][r] + acc[mt][5][r];
            }
        }
    }
}

extern "C" void kernel_launch(void* const* d_in, const int* in_sizes, int n_in,
                              void* d_out, int out_size, void* d_ws, size_t ws_size,
                              hipStream_t stream) {
    const float* x1 = (const float*)d_in[0];
    const float* x2 = (const float*)d_in[1];
    const float* w  = (const float*)d_in[2];
    float* out = (float*)d_out;
    const int n = in_sizes[0] / 1024;          // DIM_IN1 = 1024
    const int grid = (n + MT - 1) / MT;
    hipLaunchKernelGGL(cg_tp_wmma_f32, dim3(grid), dim3(512), 0, stream,
                       x1, x2, w, out, n);
}